// NeuralSamplerPosEmbLearnableLargeEnergy_77747497992226
// MI455X (gfx1250) — compile-verified
//
#include <hip/hip_runtime.h>
#include <hip/hip_bf16.h>
#include <math.h>

typedef __attribute__((ext_vector_type(2))) float v2f;
typedef __attribute__((ext_vector_type(8))) float v8f;

#define B_ 32
#define T_ 2048
#define D_ 128
#define L_ 512

// workspace offsets (in floats)
#define OFF_MAG   0
#define OFF_SCORE (B_*T_)
#define OFF_CUM   (2*B_*T_)
#define OFF_WSUM  (3*B_*T_)
#define OFF_SUMM  (3*B_*T_ + B_*L_)
#define OFF_MAXM  (OFF_SUMM + B_)
#define OFF_NEED  (OFF_MAXM + B_)
#define OFF_START (OFF_NEED + B_)
#define OFF_END   (OFF_START + B_*L_)

// ---------------- 1. magnitude[b,t] = sum_d exp(x[b,t,d]) ----------------
__global__ __launch_bounds__(256) void k_mag(const float* __restrict__ x,
                                             float* __restrict__ mag) {
    int lane = threadIdx.x & 31;
    int w    = threadIdx.x >> 5;
    int g    = blockIdx.x * 8 + w;              // flat b*T + t
    const float4* xv = reinterpret_cast<const float4*>(x + (size_t)g * D_);
    float4 v = xv[lane];                        // 4 contiguous d per lane (32 lanes * 4 = 128)
    float s = expf(v.x) + expf(v.y) + expf(v.z) + expf(v.w);
    for (int o = 16; o > 0; o >>= 1) s += __shfl_xor(s, o, 32);
    if (lane == 0) mag[g] = s;
}

// ---------------- 2. per-batch sum/max + need flag ----------------
__global__ __launch_bounds__(256) void k_stats(const float* __restrict__ mag,
                                               float* __restrict__ summag,
                                               float* __restrict__ maxmag,
                                               int*   __restrict__ needflag) {
    __shared__ float ss[256];
    __shared__ float sm[256];
    int b = blockIdx.x, tid = threadIdx.x;
    const float* m = mag + (size_t)b * T_;
    float s = 0.f, mx = 0.f;
    for (int j = 0; j < 8; ++j) { float v = m[tid * 8 + j]; s += v; mx = fmaxf(mx, v); }
    ss[tid] = s; sm[tid] = mx; __syncthreads();
    for (int o = 128; o > 0; o >>= 1) {
        if (tid < o) { ss[tid] += ss[tid + o]; sm[tid] = fmaxf(sm[tid], sm[tid + o]); }
        __syncthreads();
    }
    if (tid == 0) {
        summag[b] = ss[0];
        maxmag[b] = sm[0];
        float maxval = sm[0] * (float)L_ / ss[0];
        needflag[b] = (maxval >= 1.0f) ? 1 : 0;
    }
}

// ---------------- 3. final score + cumsum (per-batch block) ----------------
__global__ __launch_bounds__(256) void k_score(const float* __restrict__ mag,
                                               const float* __restrict__ summag,
                                               const float* __restrict__ maxmag,
                                               const int*   __restrict__ needflag,
                                               float* __restrict__ score,
                                               float* __restrict__ cum) {
    __shared__ float sd[256];
    int b = blockIdx.x, tid = threadIdx.x;
    float sumM = summag[b], maxM = maxmag[b];
    int need = needflag[b];
    int anyneed = 0;
    for (int i = 0; i < B_; ++i) anyneed |= needflag[i];

    const float* m = mag + (size_t)b * T_;
    float s2[8];
    float lsum = 0.f;
    for (int j = 0; j < 8; ++j) {
        float v  = m[tid * 8 + j];
        float s1 = v * (float)L_ / sumM;          // global max cancels analytically
        float vv = need ? (v / maxM) : s1;        // clamp by max_val when needed
        s2[j] = vv; lsum += vv;
    }
    sd[tid] = lsum; __syncthreads();
    for (int o = 128; o > 0; o >>= 1) { if (tid < o) sd[tid] += sd[tid + o]; __syncthreads(); }
    float sum2 = sd[0]; __syncthreads();
    float distance = (float)L_ - sum2;

    float iv[8]; float liv = 0.f;
    for (int j = 0; j < 8; ++j) {
        float v = s2[j];
        iv[j] = (v < 0.99f && v > 0.01f) ? (1.0f - v) : 0.0f;
        liv += iv[j];
    }
    sd[tid] = liv; __syncthreads();
    for (int o = 128; o > 0; o >>= 1) { if (tid < o) sd[tid] += sd[tid + o]; __syncthreads(); }
    float sumiv = sd[0]; __syncthreads();

    float a = (sumiv > 0.f) ? (distance / fmaxf(sumiv, 1e-12f)) : 0.f;
    a = fminf(a, 1.0f);
    int gate = (anyneed && (distance >= 1.0f)) ? 1 : 0;

    float sval[8], pref[8];
    float loc = 0.f;
    for (int j = 0; j < 8; ++j) {
        float v = s2[j] + (gate ? iv[j] * a : 0.f);
        sval[j] = v; loc += v; pref[j] = loc;
    }
    // block scan of per-thread totals (Hillis-Steele, deterministic)
    sd[tid] = loc; __syncthreads();
    for (int o = 1; o < 256; o <<= 1) {
        float t = (tid >= o) ? sd[tid - o] : 0.f;
        __syncthreads();
        sd[tid] += t;
        __syncthreads();
    }
    float excl = (tid > 0) ? sd[tid - 1] : 0.f;
    for (int j = 0; j < 8; ++j) {
        int t = tid * 8 + j;
        score[(size_t)b * T_ + t] = sval[j];
        cum  [(size_t)b * T_ + t] = excl + pref[j];
    }
}

// ---------------- 4. bucket ranges + weight sums via binary search ----------------
__device__ inline int first_greater(const float* c, float v) {
    int lo = 0, hi = T_;
    while (lo < hi) { int mid = (lo + hi) >> 1; if (c[mid] > v) hi = mid; else lo = mid + 1; }
    return lo;                                   // first t with cum[t] > v (may be T)
}

__global__ __launch_bounds__(256) void k_ranges(const float* __restrict__ cum,
                                                float* __restrict__ wsum,
                                                int*   __restrict__ startI,
                                                int*   __restrict__ endI) {
    int idx = blockIdx.x * 256 + threadIdx.x;    // b*L + l
    int b = idx / L_, l = idx % L_;
    const float* c = cum + (size_t)b * T_;
    int t0 = first_greater(c, (float)l);
    int t1 = first_greater(c, (float)(l + 1));
    startI[idx] = t0; endI[idx] = t1;
    float c0 = (t0 > 0) ? c[t0 - 1] : 0.f;
    float c1 = (t1 > 0) ? c[t1 - 1] : 0.f;
    wsum[idx] = c1 - c0;                         // deterministic bucket weight sum
}

// ---------------- 5. segmented einsum via V_WMMA_F32_16X16X4_F32 ----------------
__global__ __launch_bounds__(32) void k_wmma(const float* __restrict__ x,
                                             const float* __restrict__ pos,
                                             const float* __restrict__ score,
                                             const float* __restrict__ cum,
                                             const float* __restrict__ wsum,
                                             const int*   __restrict__ startI,
                                             const int*   __restrict__ endI,
                                             float* __restrict__ out) {
    int bid = blockIdx.x;
    int b   = bid >> 5;                 // / (L/16) == 32 tiles per batch
    int lt  = bid & 31;
    int l0  = lt * 16;
    int ts  = startI[b * L_ + l0];
    int te  = endI  [b * L_ + l0 + 15];

    int lane = threadIdx.x;
    int half = lane >> 4;               // 0: K=0,1   1: K=2,3  (A 16x4 f32 layout)
    int lidx = lane & 15;               // M for A, N for B/C
    int kb   = half * 2;

    const float* xb = x     + (size_t)b * T_ * D_;
    const float* cb = cum   + (size_t)b * T_;
    const float* sb = score + (size_t)b * T_;

    v8f z = {};
    v8f accF[8], accE[8];
#pragma unroll
    for (int i = 0; i < 8; ++i) { accF[i] = z; accE[i] = z; }

    for (int k = ts; k < te; k += 4) {
        int t0 = k + kb, t1 = t0 + 1;
        // clamp indices so every load is unconditional & in-bounds; padded
        // columns contribute nothing because A is zeroed branchlessly below.
        int tc0 = (t0 < T_ - 1) ? t0 : (T_ - 1);
        int tc1 = (t1 < T_ - 1) ? t1 : (T_ - 1);
        float s0 = sb[tc0], s1 = sb[tc1];
        float c0 = cb[tc0], c1 = cb[tc1];
        int bk0 = (int)ceilf(c0) - 1;            // bucket of t0
        int bk1 = (int)ceilf(c1) - 1;
        v2f a;
        a.x = ((t0 < te) && (bk0 == l0 + lidx)) ? s0 : 0.f;
        a.y = ((t1 < te) && (bk1 == l0 + lidx)) ? s1 : 0.f;

        const float* xr0 = xb  + (size_t)tc0 * D_ + lidx;
        const float* xr1 = xb  + (size_t)tc1 * D_ + lidx;
        const float* pr0 = pos + (size_t)tc0 * D_ + lidx;
        const float* pr1 = pos + (size_t)tc1 * D_ + lidx;

        // speculative prefetch of next chunk's x rows (in-bounds, branchless)
        int tpf = k + 4 + kb; tpf = (tpf < T_ - 1) ? tpf : (T_ - 1);
        __builtin_prefetch(xb + (size_t)tpf * D_ + lidx * 8, 0, 1);

#pragma unroll
        for (int dt = 0; dt < 8; ++dt) {
            v2f bx, bp;
            bx.x = xr0[dt * 16];
            bx.y = xr1[dt * 16];
            bp.x = pr0[dt * 16];
            bp.y = pr1[dt * 16];
            accF[dt] = __builtin_amdgcn_wmma_f32_16x16x4_f32(false, a, false, bx, (short)0, accF[dt], false, false);
            accE[dt] = __builtin_amdgcn_wmma_f32_16x16x4_f32(false, a, false, bp, (short)0, accE[dt], false, false);
        }
    }

    // C layout: VGPR i -> M = i + 8*half, N = lidx
#pragma unroll
    for (int dt = 0; dt < 8; ++dt) {
        int d0 = dt * 16;
#pragma unroll
        for (int i = 0; i < 8; ++i) {
            int l = l0 + i + half * 8;
            int d = d0 + lidx;
            float w = wsum[b * L_ + l] + 1e-8f;
            out[(((size_t)b * 2 + 0) * L_ + l) * D_ + d] = accF[dt][i] / w;   // feat
            out[(((size_t)b * 2 + 1) * L_ + l) * D_ + d] = accE[dt][i] / w;   // emb
        }
    }
}

extern "C" void kernel_launch(void* const* d_in, const int* in_sizes, int n_in,
                              void* d_out, int out_size, void* d_ws, size_t ws_size,
                              hipStream_t stream) {
    const float* x   = (const float*)d_in[0];   // [B,T,D]
    const float* pos = (const float*)d_in[1];   // [1,T,D]
    float* out = (float*)d_out;                 // [B,2,L,D]

    float* f       = (float*)d_ws;
    float* mag     = f + OFF_MAG;
    float* score   = f + OFF_SCORE;
    float* cum     = f + OFF_CUM;
    float* wsum    = f + OFF_WSUM;
    float* summag  = f + OFF_SUMM;
    float* maxmag  = f + OFF_MAXM;
    int*   needflg = (int*)(f + OFF_NEED);
    int*   startI  = (int*)(f + OFF_START);
    int*   endI    = (int*)(f + OFF_END);

    k_mag   <<<B_ * T_ / 8,   256, 0, stream>>>(x, mag);
    k_stats <<<B_,            256, 0, stream>>>(mag, summag, maxmag, needflg);
    k_score <<<B_,            256, 0, stream>>>(mag, summag, maxmag, needflg, score, cum);
    k_ranges<<<B_ * L_ / 256, 256, 0, stream>>>(cum, wsum, startI, endI);
    k_wmma  <<<B_ * (L_ / 16), 32, 0, stream>>>(x, pos, score, cum, wsum, startI, endI, out);
}